// Graph_diffusion_12867722019642
// MI455X (gfx1250) — compile-verified
//
#include <hip/hip_runtime.h>

typedef __attribute__((ext_vector_type(2))) float v2f;
typedef __attribute__((ext_vector_type(8))) float v8f;

#define NEDGETYPE 5
#define NNODETYPE 7
#define CIN       4
#define XHC       11     // CIN + NNODETYPE
#define KDIM      55     // NEDGETYPE * XHC
#define KPAD      56     // padded K (multiple of 4, even -> 8B-aligned float2 loads)
#define COUT      32
#define KSTEPS    14     // KPAD / 4

// ---------------------------------------------------------------- zero agg
__global__ void gd_zero_f4(float4* __restrict__ p, long n4) {
  long i = (long)blockIdx.x * blockDim.x + threadIdx.x;
  if (i < n4) p[i] = make_float4(0.f, 0.f, 0.f, 0.f);
}

// ---------------------------------------------------- time embedding -> inj
// B<=8 rows: sinusoid -> lin1 -> swish -> lin2 ; inj = swish(temb) @ W_temb
__global__ void __launch_bounds__(512)
gd_temb_inj(const float* __restrict__ t,  const float* __restrict__ W1,
            const float* __restrict__ b1, const float* __restrict__ W2,
            const float* __restrict__ b2, const float* __restrict__ Wt,
            float* __restrict__ inj, int B) {
  __shared__ float emb[8 * 128];
  __shared__ float h1 [8 * 512];
  __shared__ float t2 [8 * 512];
  const int tid = threadIdx.x;

  for (int i = tid; i < B * 128; i += 512) {
    int b = i >> 7, j = i & 127;
    // freqs = exp(arange(64) * (-ln(10000)/63))
    float freq = expf(-(9.210340371976184f / 63.0f) * (float)(j & 63));
    float ang  = t[b] * freq;
    emb[i] = (j < 64) ? sinf(ang) : cosf(ang);
  }
  __syncthreads();

  const int c = tid;                       // 0..511: one output column
  for (int b = 0; b < B; ++b) {
    float s = b1[c];
    for (int k = 0; k < 128; ++k) s = fmaf(emb[b * 128 + k], W1[k * 512 + c], s);
    h1[b * 512 + c] = s / (1.0f + expf(-s));            // swish
  }
  __syncthreads();

  for (int b = 0; b < B; ++b) {
    float s = b2[c];
    for (int k = 0; k < 512; ++k) s = fmaf(h1[b * 512 + k], W2[k * 512 + c], s);
    t2[b * 512 + c] = s / (1.0f + expf(-s));            // swish(temb)
  }
  __syncthreads();

  if (c < B * COUT) {
    int b = c >> 5, n = c & 31;
    float s = 0.f;
    for (int k = 0; k < 512; ++k) s = fmaf(t2[b * 512 + k], Wt[k * COUT + n], s);
    inj[c] = s;
  }
}

// ------------------------------------------------------------ edge scatter
// 5 lanes per edge: lanes 0-3 add x channels, lane 4 adds the one-hot hit.
// agg layout: [N, KPAD] with columns et*11 + c (c<11); cols 55 stay zero.
__global__ void gd_scatter(const float* __restrict__ x,
                           const int* __restrict__ ei,
                           const int* __restrict__ etyp,
                           const int* __restrict__ ntyp,
                           float* __restrict__ agg, int E) {
  long tid = (long)blockIdx.x * blockDim.x + threadIdx.x;
  long e = tid / 5;
  int  c = (int)(tid - e * 5);
  if (e >= E) return;
  int dst = ei[e];
  int src = ei[(long)E + e];
  int et  = etyp[e];
  float v; int pos;
  if (c < 4) { v = x[(long)src * CIN + c]; pos = et * XHC + c; }
  else       { v = 1.0f;                   pos = et * XHC + CIN + ntyp[src]; }
  atomicAdd(agg + (long)dst * KPAD + pos, v);
}

// ----------------------------------------------------- WMMA GEMM + epilogue
// out[n, 0:32] = (agg[n, 0:55] @ W_conv) * 0.2 + inj[batch_id[n]]
// One wave -> one 16-node tile as two v_wmma_f32_16x16x4_f32 accumulators.
__global__ void __launch_bounds__(256)
gd_gemm_wmma(const float* __restrict__ agg, const float* __restrict__ Wc,
             const int* __restrict__ batch_id, const float* __restrict__ inj,
             float* __restrict__ out, int N, int numTiles) {
  __shared__ float sW[KPAD * COUT];                       // zero-padded W_conv
  for (int i = threadIdx.x; i < KPAD * COUT; i += 256)
    sW[i] = (i < KDIM * COUT) ? Wc[i] : 0.0f;
  __syncthreads();

  const int lane = threadIdx.x & 31;
  const int wave = threadIdx.x >> 5;
  const int g    = lane >> 4;                             // K half-select
  const int l    = lane & 15;                             // M (A) / N (B,C,D)

  // Preload B fragments for the whole K loop (reused across node tiles).
  // B 4x16 layout: lanes 0-15 hold K = {k, k+1}, lanes 16-31 hold {k+2, k+3}.
  v2f Bf0[KSTEPS], Bf1[KSTEPS];
#pragma unroll
  for (int kk = 0; kk < KSTEPS; ++kk) {
    int kb = kk * 4 + 2 * g;
    Bf0[kk].x = sW[kb * COUT + l];
    Bf0[kk].y = sW[(kb + 1) * COUT + l];
    Bf1[kk].x = sW[kb * COUT + 16 + l];
    Bf1[kk].y = sW[(kb + 1) * COUT + 16 + l];
  }

  // Grid-strided tiles; condition is wave-uniform so EXEC stays all-ones.
  for (int tile = blockIdx.x * 8 + wave; tile < numTiles; tile += gridDim.x * 8) {
    const int base = tile * 16;
    int arow = base + l; if (arow > N - 1) arow = N - 1;  // clamp (N%16==0 normally)
    const float* Ar = agg + (long)arow * KPAD + 2 * g;    // A: M=l, K = k + 2g + {0,1}

    v8f acc0 = {}; v8f acc1 = {};
#pragma unroll
    for (int kk = 0; kk < KSTEPS; ++kk) {
      v2f a = *(const v2f*)(Ar + kk * 4);                 // 8B-aligned
      acc0 = __builtin_amdgcn_wmma_f32_16x16x4_f32(
          false, a, false, Bf0[kk], (short)0, acc0, false, false);
      acc1 = __builtin_amdgcn_wmma_f32_16x16x4_f32(
          false, a, false, Bf1[kk], (short)0, acc1, false, false);
    }

    // C/D layout: VGPR r -> M = r + 8g, N = l (+16 for second tile)
#pragma unroll
    for (int r = 0; r < 8; ++r) {
      int row = base + r + 8 * g;
      if (row < N) {
        int bb = batch_id[row];
        out[(long)row * COUT + l]      = acc0[r] * 0.2f + inj[bb * COUT + l];
        out[(long)row * COUT + 16 + l] = acc1[r] * 0.2f + inj[bb * COUT + 16 + l];
      }
    }
  }
}

// ------------------------------------------------------------------- launch
extern "C" void kernel_launch(void* const* d_in, const int* in_sizes, int n_in,
                              void* d_out, int out_size, void* d_ws, size_t ws_size,
                              hipStream_t stream) {
  const float* x    = (const float*)d_in[0];
  const float* t    = (const float*)d_in[1];
  const int*   ei   = (const int*)d_in[2];
  const int*   etyp = (const int*)d_in[3];
  const int*   ntyp = (const int*)d_in[4];
  const int*   bid  = (const int*)d_in[5];
  const float* Wc   = (const float*)d_in[6];
  const float* W1   = (const float*)d_in[7];
  const float* b1   = (const float*)d_in[8];
  const float* W2   = (const float*)d_in[9];
  const float* b2   = (const float*)d_in[10];
  const float* Wt   = (const float*)d_in[11];

  const int N = in_sizes[4];         // node_type count
  const int E = in_sizes[3];         // edge_type count
  int B = in_sizes[1]; if (B > 8) B = 8;

  float* agg = (float*)d_ws;                       // [N, KPAD]  (~89.6 MB, L2-resident)
  float* inj = agg + (size_t)N * KPAD;             // [B, 32]
  float* out = (float*)d_out;

  long n4 = ((long)N * KPAD) / 4;                  // KPAD=56 -> always divisible
  gd_zero_f4<<<(int)((n4 + 255) / 256), 256, 0, stream>>>((float4*)agg, n4);

  gd_temb_inj<<<1, 512, 0, stream>>>(t, W1, b1, W2, b2, Wt, inj, B);

  long nth = (long)E * 5;
  gd_scatter<<<(int)((nth + 255) / 256), 256, 0, stream>>>(x, ei, etyp, ntyp, agg, E);

  int numTiles = (N + 15) / 16;
  int blocks = (numTiles + 7) / 8; if (blocks > 1536) blocks = 1536;
  gd_gemm_wmma<<<blocks, 256, 0, stream>>>(agg, Wc, bid, inj, out, N, numTiles);
}